// SAGE_Body_59846074302988
// MI455X (gfx1250) — compile-verified
//
#include <hip/hip_runtime.h>
#include <hip/hip_bf16.h>

#define N_NODES   100000
#define NFEAT     128
#define N_EDGES   1600000
#define EPS_NORM  1e-12f
#define EPS_BN    1e-5f

typedef __attribute__((ext_vector_type(2))) float v2f;
typedef __attribute__((ext_vector_type(8))) float v8f;

// ---------------------------------------------------------------------------
// Edge scatter: agg[dst,:] += feat[src,:]. One wave32 per edge, 4 feats/lane.
// agg (51MB) is L2-resident on MI455X (192MB L2) so atomics stay on-chip.
// First pass also accumulates the destination degree (lane 0).
// ---------------------------------------------------------------------------
template <bool COUNT>
__global__ __launch_bounds__(256)
void scatter_kernel(const float* __restrict__ feat,
                    const long long* __restrict__ ei,
                    float* __restrict__ agg,
                    float* __restrict__ cnt) {
    const int wave = threadIdx.x >> 5;
    const int lane = threadIdx.x & 31;
    long long e = (long long)blockIdx.x * 8 + wave;
    if (e >= N_EDGES) return;
    const int src = (int)ei[e];
    const int dst = (int)ei[N_EDGES + e];
    const float4 v = *(const float4*)&feat[(size_t)src * NFEAT + lane * 4];
    float* ap = &agg[(size_t)dst * NFEAT + lane * 4];
    atomicAdd(ap + 0, v.x);
    atomicAdd(ap + 1, v.y);
    atomicAdd(ap + 2, v.z);
    atomicAdd(ap + 3, v.w);
    if (COUNT && lane == 0) atomicAdd(&cnt[dst], 1.0f);
}

// ---------------------------------------------------------------------------
// Fused SAGE layer: out[16 rows x 128 cols per block]
//   C = (agg @ Wl^T) * inv_deg_row + x @ Wr^T + bl
//   (pass-1 row scaling hoisted out of the WMMA loop: inv depends only on the
//    output ROW, so scale the accumulator once between the two passes)
//   row-wise L2 normalize (cross-wave LDS reduction)
//   LAYER1: + ReLU, write h, accumulate BN per-feature sum / sumsq
// Wave w owns output columns [16w,16w+16); weights staged per-wave in LDS.
// ---------------------------------------------------------------------------
template <bool LAYER1>
__global__ __launch_bounds__(256)
void sage_gemm_kernel(const float* __restrict__ agg,
                      const float* __restrict__ cnt,
                      const float* __restrict__ xin,
                      const float* __restrict__ Wl,
                      const float* __restrict__ bl,
                      const float* __restrict__ Wr,
                      float* __restrict__ out,
                      float* __restrict__ bnsum,
                      float* __restrict__ bnsumsq) {
    __shared__ float Bs[8 * 16 * NFEAT];  // 64 KB: 16 weight columns per wave

    const int tid  = threadIdx.x;
    const int wave = tid >> 5;
    const int lane = tid & 31;
    const int n    = lane & 15;       // output column within tile / A row sel
    const int hgrp = lane >> 4;       // half-wave group
    const int row0 = blockIdx.x * 16;
    const int n0   = wave * 16;
    const int wso  = wave * (16 * NFEAT);
    const int r    = row0 + n;        // A-matrix row this lane feeds
    const int koff = 2 * hgrp;        // K-pair offset per ISA A/B layout

    // ---- stage Wl columns [n0, n0+16) ----
    for (int t = lane; t < 512; t += 32) {
        const int idx = t * 4;
        const int rr = idx >> 7, cc = idx & 127;
        *(float4*)&Bs[wso + idx] = *(const float4*)&Wl[(n0 + rr) * NFEAT + cc];
    }
    __syncthreads();

    v8f c = {0.f, 0.f, 0.f, 0.f, 0.f, 0.f, 0.f, 0.f};

    // ---- pass 1: raw neighbor sums @ Wl^T (mean scaling deferred) ----
    const float inv = 1.0f / fmaxf(cnt[r], 1.0f);   // lane n holds row n's inv
    const float* __restrict__ arow = agg + (size_t)r * NFEAT;
#pragma unroll
    for (int k = 0; k < NFEAT; k += 4) {
        v2f a = *(const v2f*)&arow[k + koff];
        v2f b = *(const v2f*)&Bs[wso + n * NFEAT + k + koff];
        c = __builtin_amdgcn_wmma_f32_16x16x4_f32(false, a, false, b,
                                                  (short)0, c, false, false);
    }

    // ---- apply per-row 1/deg to the accumulator (8 shfl + 8 mul) ----
#pragma unroll
    for (int v = 0; v < 8; ++v) {
        const int m = v + 8 * hgrp;           // C/D row held in c[v]
        c[v] *= __shfl(inv, m, 32);           // broadcast row m's inverse
    }

    __syncthreads();
    // ---- stage Wr columns ----
    for (int t = lane; t < 512; t += 32) {
        const int idx = t * 4;
        const int rr = idx >> 7, cc = idx & 127;
        *(float4*)&Bs[wso + idx] = *(const float4*)&Wr[(n0 + rr) * NFEAT + cc];
    }
    __syncthreads();

    // ---- pass 2: self features @ Wr^T ----
    const float* __restrict__ xrow = xin + (size_t)r * NFEAT;
#pragma unroll
    for (int k = 0; k < NFEAT; k += 4) {
        v2f a = *(const v2f*)&xrow[k + koff];
        v2f b = *(const v2f*)&Bs[wso + n * NFEAT + k + koff];
        c = __builtin_amdgcn_wmma_f32_16x16x4_f32(false, a, false, b,
                                                  (short)0, c, false, false);
    }

    // ---- bias (per output column) ----
    const float bias = bl[n0 + n];
#pragma unroll
    for (int v = 0; v < 8; ++v) c[v] += bias;

    // ---- row-wise sum of squares across all 8 waves (reuse LDS) ----
    __syncthreads();               // all waves done reading Bs
    if (tid < 16) Bs[tid] = 0.0f;  // Bs[m] = sumsq of output row m
    __syncthreads();
#pragma unroll
    for (int v = 0; v < 8; ++v)
        atomicAdd(&Bs[v + 8 * hgrp], c[v] * c[v]);
    __syncthreads();

    // ---- normalize, (ReLU), store, (BN partial sums) ----
    float s1 = 0.f, s2 = 0.f;
#pragma unroll
    for (int v = 0; v < 8; ++v) {
        const int m = v + 8 * hgrp;                 // C/D row per ISA layout
        const float nrm = sqrtf(Bs[m]);
        float val = c[v] / fmaxf(nrm, EPS_NORM);
        if (LAYER1) val = fmaxf(val, 0.0f);
        out[(size_t)(row0 + m) * NFEAT + n0 + n] = val;
        if (LAYER1) { s1 += val; s2 += val * val; }
    }
    if (LAYER1) {
        s1 += __shfl_xor(s1, 16, 32);  // lanes L and L+16 share column n0+n
        s2 += __shfl_xor(s2, 16, 32);
        if (hgrp == 0) {
            atomicAdd(&bnsum[n0 + n], s1);
            atomicAdd(&bnsumsq[n0 + n], s2);
        }
    }
}

// ---------------------------------------------------------------------------
// BN finalize: fold batch stats + gamma/beta into per-feature scale/shift
// ---------------------------------------------------------------------------
__global__ __launch_bounds__(128)
void bn_finalize_kernel(const float* __restrict__ bnsum,
                        const float* __restrict__ bnsumsq,
                        const float* __restrict__ gamma,
                        const float* __restrict__ beta,
                        float* __restrict__ scale,
                        float* __restrict__ shift) {
    const int j = threadIdx.x;
    const float invN = 1.0f / (float)N_NODES;
    const float mu  = bnsum[j] * invN;
    const float var = bnsumsq[j] * invN - mu * mu;
    const float s   = gamma[j] * rsqrtf(var + EPS_BN);
    scale[j] = s;
    shift[j] = beta[j] - mu * s;
}

// ---------------------------------------------------------------------------
// BN apply in place on h (float4 vectorized)
// ---------------------------------------------------------------------------
__global__ __launch_bounds__(256)
void bn_apply_kernel(float* __restrict__ h,
                     const float* __restrict__ scale,
                     const float* __restrict__ shift) {
    const long long total4 = (long long)N_NODES * NFEAT / 4;
    long long t = (long long)blockIdx.x * blockDim.x + threadIdx.x;
    if (t >= total4) return;
    const int j0 = (int)((t * 4) & (NFEAT - 1));
    float4 v = *(float4*)&h[t * 4];
    v.x = v.x * scale[j0 + 0] + shift[j0 + 0];
    v.y = v.y * scale[j0 + 1] + shift[j0 + 1];
    v.z = v.z * scale[j0 + 2] + shift[j0 + 2];
    v.w = v.w * scale[j0 + 3] + shift[j0 + 3];
    *(float4*)&h[t * 4] = v;
}

// ---------------------------------------------------------------------------
extern "C" void kernel_launch(void* const* d_in, const int* in_sizes, int n_in,
                              void* d_out, int out_size, void* d_ws, size_t ws_size,
                              hipStream_t stream) {
    const float*     x     = (const float*)d_in[0];
    const long long* ei    = (const long long*)d_in[1];  // int64 [2, E]
    const float*     Wl1   = (const float*)d_in[2];
    const float*     bl1   = (const float*)d_in[3];
    const float*     Wr1   = (const float*)d_in[4];
    const float*     gamma = (const float*)d_in[5];
    const float*     beta  = (const float*)d_in[6];
    const float*     Wl2   = (const float*)d_in[7];
    const float*     bl2   = (const float*)d_in[8];
    const float*     Wr2   = (const float*)d_in[9];
    float*           out   = (float*)d_out;

    const size_t NB = (size_t)N_NODES * NFEAT * sizeof(float);  // 51.2 MB
    char* ws = (char*)d_ws;
    float* agg     = (float*)(ws);                   // reused for both layers
    float* h       = (float*)(ws + NB);              // BN applied in place
    float* cnt     = (float*)(ws + 2 * NB);
    char*  stats   = ws + 2 * NB + (size_t)N_NODES * sizeof(float);
    float* bnsum   = (float*)(stats);
    float* bnsumsq = (float*)(stats + 512);
    float* bnscale = (float*)(stats + 1024);
    float* bnshift = (float*)(stats + 1536);

    const int scatterGrid = (N_EDGES + 7) / 8;           // one wave per edge
    const int gemmGrid    = N_NODES / 16;                // 6250, exact
    const int bnGrid      = (N_NODES * NFEAT / 4 + 255) / 256;

    // zero accumulators (scratch persists across graph replays)
    hipMemsetAsync(agg, 0, NB, stream);
    hipMemsetAsync(cnt, 0, (size_t)N_NODES * sizeof(float), stream);
    hipMemsetAsync(bnsum, 0, NFEAT * sizeof(float), stream);
    hipMemsetAsync(bnsumsq, 0, NFEAT * sizeof(float), stream);

    // ---- layer 1 (scatter also accumulates degrees) ----
    scatter_kernel<true><<<scatterGrid, 256, 0, stream>>>(x, ei, agg, cnt);
    sage_gemm_kernel<true><<<gemmGrid, 256, 0, stream>>>(
        agg, cnt, x, Wl1, bl1, Wr1, h, bnsum, bnsumsq);

    // ---- transition: ReLU folded above; BN here ----
    bn_finalize_kernel<<<1, 128, 0, stream>>>(bnsum, bnsumsq, gamma, beta,
                                              bnscale, bnshift);
    bn_apply_kernel<<<bnGrid, 256, 0, stream>>>(h, bnscale, bnshift);

    // ---- layer 2 (reuse agg buffer; degrees unchanged) ----
    hipMemsetAsync(agg, 0, NB, stream);
    scatter_kernel<false><<<scatterGrid, 256, 0, stream>>>(h, ei, agg, nullptr);
    sage_gemm_kernel<false><<<gemmGrid, 256, 0, stream>>>(
        agg, cnt, h, Wl2, bl2, Wr2, out, nullptr, nullptr);
}